// PointLaplacianLoss_26628797235302
// MI455X (gfx1250) — compile-verified
//
#include <hip/hip_runtime.h>
#include <hip/hip_bf16.h>
#include <float.h>

typedef float v2f __attribute__((ext_vector_type(2)));
typedef float v4f __attribute__((ext_vector_type(4)));
typedef float v8f __attribute__((ext_vector_type(8)));

#define NB     2
#define NPTS   8192
#define NDIM   3
#define KNN    10
#define TILE   16
#define NTILES (NPTS / TILE)   // 512

// ---------------------------------------------------------------------------
// Kernel 1: kNN via WMMA F32 16x16x4 distance tiles + per-lane top-10.
// One wave (32 threads) per 16-row tile of the distance matrix.
// D = (-2A) x B + C, with C[m][n] = |p_m|^2 + |p_n|^2  ==> D = squared dist.
// B-operand loads are software-pipelined one tile ahead to hide L2 latency
// (only 1024 waves total -> must hide latency with ILP, not occupancy).
// ---------------------------------------------------------------------------
__global__ __launch_bounds__(32)
void PointLap_knn_kernel(const float* __restrict__ P, int* __restrict__ knn) {
    const int blk = blockIdx.x;
    const int b   = blk / NTILES;
    const int i0  = (blk % NTILES) * TILE;
    const int l   = threadIdx.x;     // 0..31
    const int m   = l & 15;
    const int hi  = l >> 4;          // 0 or 1
    const float* Pb = P + (size_t)b * NPTS * NDIM;

    __shared__ float tile[TILE * TILE];
    __shared__ float md[32 * KNN];
    __shared__ int   mi[32 * KNN];

    // A matrix 16x4 (K padded: x,y,z,0), pre-scaled by -2.
    // Layout: lanes 0-15 hold K=0(v0),K=1(v1) for M=lane; lanes 16-31 K=2,K=3.
    const int rowA = i0 + m;
    v2f a;
    if (!hi) { a.x = -2.0f * Pb[rowA*3+0]; a.y = -2.0f * Pb[rowA*3+1]; }
    else     { a.x = -2.0f * Pb[rowA*3+2]; a.y = 0.0f; }

    // |p|^2 for the 8 rows this lane's C/D VGPRs correspond to (M = v + 8*hi).
    float sqI[8];
#pragma unroll
    for (int v = 0; v < 8; ++v) {
        const int r = i0 + v + 8*hi;
        const float x = Pb[r*3+0], y = Pb[r*3+1], z = Pb[r*3+2];
        sqI[v] = x*x + y*y + z*z;
    }

    // Per-lane top-10 (sorted ascending; bd[9] is current worst).
    float bd[KNN]; int bi[KNN];
#pragma unroll
    for (int t = 0; t < KNN; ++t) { bd[t] = FLT_MAX; bi[t] = 0; }

    // Preload tile 0's B-point coordinates.
    float nx = Pb[m*3+0], ny = Pb[m*3+1], nz = Pb[m*3+2];

    for (int j0 = 0; j0 < NPTS; j0 += TILE) {
        // Consume the preloaded coords for this tile.
        const float x = nx, y = ny, z = nz;

        // Issue next tile's loads early (software pipeline); uniform branch.
        const int jn = j0 + TILE;
        if (jn < NPTS) {
            const int jm = jn + m;
            nx = Pb[jm*3+0]; ny = Pb[jm*3+1]; nz = Pb[jm*3+2];
        }
        // Pre-warm WGP cache a few tiles ahead (global_prefetch_b8).
        if (j0 + 4*TILE < NPTS) {
            __builtin_prefetch(&Pb[(j0 + 4*TILE + m)*3], 0, 1);
        }

        const float sqj = x*x + y*y + z*z;

        // B matrix 4x16: lanes 0-15 hold K=0(v0),K=1(v1) for N=lane;
        // lanes 16-31 hold K=2(v0),K=3(v1).  B[k][n] = P[j0+n][k].
        v2f bv;
        bv.x = hi ? z : x;
        bv.y = hi ? 0.0f : y;

        v8f c;
#pragma unroll
        for (int v = 0; v < 8; ++v) c[v] = sqI[v] + sqj;

        // (neg_a, A, neg_b, B, c_mod, C, reuse_a, reuse_b)
        v8f d = __builtin_amdgcn_wmma_f32_16x16x4_f32(
            false, a, false, bv, (short)0, c, false, false);

        // Scatter the 16x16 distance tile to LDS (row-major).
#pragma unroll
        for (int v = 0; v < 8; ++v) tile[(v + 8*hi)*TILE + m] = d[v];
        __syncthreads();

        // Each lane scans half of row r = m (cols cb..cb+7); the 8 floats are
        // contiguous and 32B aligned -> two ds_load_b128.
        const int cb = hi * 8;
        const v4f dv0 = *(const v4f*)&tile[m*TILE + cb + 0];
        const v4f dv1 = *(const v4f*)&tile[m*TILE + cb + 4];
        float dcand[8] = { dv0.x, dv0.y, dv0.z, dv0.w, dv1.x, dv1.y, dv1.z, dv1.w };
#pragma unroll
        for (int t = 0; t < 8; ++t) {
            const float dd  = dcand[t];
            const int   idx = j0 + cb + t;
            if (idx != (i0 + m) && dd < bd[KNN-1]) {
                bd[KNN-1] = dd; bi[KNN-1] = idx;
#pragma unroll
                for (int s = KNN-1; s > 0; --s) {
                    if (bd[s] < bd[s-1]) {
                        const float tf = bd[s]; bd[s] = bd[s-1]; bd[s-1] = tf;
                        const int   ti = bi[s]; bi[s] = bi[s-1]; bi[s-1] = ti;
                    }
                }
            }
        }
        __syncthreads();
    }

    // Merge the two half-row lists (lane l and lane l+16 share row l, l<16).
#pragma unroll
    for (int t = 0; t < KNN; ++t) { md[l*KNN + t] = bd[t]; mi[l*KNN + t] = bi[t]; }
    __syncthreads();
    if (l < 16) {
        int pa = 0, pb = 0;
        int* out = knn + ((size_t)b * NPTS + (i0 + l)) * KNN;
#pragma unroll
        for (int t = 0; t < KNN; ++t) {
            const float da = md[l*KNN + pa];
            const float db = md[(l+16)*KNN + pb];
            if (da <= db) { out[t] = mi[l*KNN + pa];      ++pa; }
            else          { out[t] = mi[(l+16)*KNN + pb]; ++pb; }
        }
    }
}

// ---------------------------------------------------------------------------
// Kernel 2: per-point |lap1 - lap2| L1 sum over D, block tree-reduce -> partial
// ---------------------------------------------------------------------------
__global__ __launch_bounds__(256)
void PointLap_loss_kernel(const float* __restrict__ P1, const float* __restrict__ P2,
                          const int* __restrict__ knn, float* __restrict__ partial) {
    __shared__ float red[256];
    const int t = blockIdx.x * 256 + threadIdx.x;   // 0..16383
    float s = 0.0f;
    if (t < NB * NPTS) {
        const int b = t / NPTS;
        const int i = t % NPTS;
        const float* p1 = P1 + (size_t)b * NPTS * NDIM;
        const float* p2 = P2 + (size_t)b * NPTS * NDIM;
        const int*   kn = knn + (size_t)t * KNN;
        float m1x=0.f,m1y=0.f,m1z=0.f, m2x=0.f,m2y=0.f,m2z=0.f;
#pragma unroll
        for (int q = 0; q < KNN; ++q) {
            const int j = kn[q];
            m1x += p1[j*3+0]; m1y += p1[j*3+1]; m1z += p1[j*3+2];
            m2x += p2[j*3+0]; m2y += p2[j*3+1]; m2z += p2[j*3+2];
        }
        const float inv = 1.0f / (float)KNN;
        const float l1x = m1x*inv - p1[i*3+0], l2x = m2x*inv - p2[i*3+0];
        const float l1y = m1y*inv - p1[i*3+1], l2y = m2y*inv - p2[i*3+1];
        const float l1z = m1z*inv - p1[i*3+2], l2z = m2z*inv - p2[i*3+2];
        s = fabsf(l1x - l2x) + fabsf(l1y - l2y) + fabsf(l1z - l2z);
    }
    red[threadIdx.x] = s;
    __syncthreads();
    for (int off = 128; off > 0; off >>= 1) {
        if (threadIdx.x < off) red[threadIdx.x] += red[threadIdx.x + off];
        __syncthreads();
    }
    if (threadIdx.x == 0) partial[blockIdx.x] = red[0];
}

// ---------------------------------------------------------------------------
// Kernel 3: reduce 64 block partials deterministically, scale to mean.
// ---------------------------------------------------------------------------
__global__ __launch_bounds__(64)
void PointLap_final_kernel(const float* __restrict__ partial, float* __restrict__ out) {
    __shared__ float red[64];
    red[threadIdx.x] = partial[threadIdx.x];
    __syncthreads();
    for (int off = 32; off > 0; off >>= 1) {
        if (threadIdx.x < off) red[threadIdx.x] += red[threadIdx.x + off];
        __syncthreads();
    }
    if (threadIdx.x == 0)
        out[0] = red[0] * (1.0f / (float)(NB * NPTS * NDIM));
}

extern "C" void kernel_launch(void* const* d_in, const int* in_sizes, int n_in,
                              void* d_out, int out_size, void* d_ws, size_t ws_size,
                              hipStream_t stream) {
    const float* point1 = (const float*)d_in[0];
    const float* point2 = (const float*)d_in[1];
    float* out = (float*)d_out;

    // Workspace layout: [ knn indices : NB*NPTS*KNN ints ][ 64 partial floats ]
    int*   knn     = (int*)d_ws;
    float* partial = (float*)((char*)d_ws + (size_t)NB * NPTS * KNN * sizeof(int));

    // 1) kNN graph from point1 (WMMA distance tiles + top-10 selection).
    PointLap_knn_kernel<<<NB * NTILES, 32, 0, stream>>>(point1, knn);

    // 2) per-point Laplacian L1 sums + block partials (64 blocks of 256).
    PointLap_loss_kernel<<<(NB * NPTS) / 256, 256, 0, stream>>>(point1, point2, knn, partial);

    // 3) deterministic final reduction -> scalar mean.
    PointLap_final_kernel<<<1, 64, 0, stream>>>(partial, out);
}